// ArgKeyFactIndex_66786741453003
// MI455X (gfx1250) — compile-verified
//
#include <hip/hip_runtime.h>
#include <hip/hip_bf16.h>

#define CNO 10000
#define PAD 10001
#define KS  10003   // max(CNO, PAD) + 2

typedef int v2i __attribute__((ext_vector_type(2)));  // clang vector: OK for NT builtin

// One wave32 per query row. The row index is forced into an SGPR via
// readfirstlane so the whole query decode / table select / segment-bound
// lookup compiles to SMEM + SALU (wave-uniform front-end), and only the
// contiguous order[] gather + NT output stores use the vector path.
__global__ __launch_bounds__(256) void ArgKeyFactIndex_gather_kernel(
    const int* __restrict__ q,                                   // B x 3
    const int* __restrict__ o0, const int* __restrict__ s0, const int* __restrict__ l0,
    const int* __restrict__ o1, const int* __restrict__ s1, const int* __restrict__ l1,
    const int* __restrict__ op, const int* __restrict__ sp, const int* __restrict__ lp,
    const int* __restrict__ mrp,                                 // max_results (device scalar)
    int* __restrict__ out,                                       // [B*MR fact][B*MR valid]
    int B, int F, int n0, int n1, int np)
{
    const int MR   = mrp[0];                    // 64 in the reference setup
    const int lane = threadIdx.x & 31;
    // Wave-uniform row index in an SGPR -> scalar front-end.
    const int wave = __builtin_amdgcn_readfirstlane(threadIdx.x >> 5);
    const int row  = blockIdx.x * 8 + wave;
    if (row >= B) return;

    // Scalar (SMEM) query decode.
    const int pred = q[3 * row + 0];
    const int a0   = q[3 * row + 1];
    const int a1   = q[3 * row + 2];

    const bool c0 = (a0 <= CNO) && (a0 != PAD);
    const bool c1 = (a1 <= CNO) && (a1 != PAD);
    const bool bv = !c0 && !c1 && (pred != PAD);

    // Table selection exactly matching the reference where() chain:
    //   both_var -> predicate table; elif is_c0 -> a0 table; else -> a1 table.
    const int* __restrict__ order;
    const int* __restrict__ starts;
    const int* __restrict__ lens;
    int key, n;
    bool vbase;
    if (bv) {
        order = op; starts = sp; lens = lp;
        key = pred;              n = np; vbase = true;
    } else if (c0) {
        order = o0; starts = s0; lens = l0;
        key = pred * KS + a0;    n = n0; vbase = true;
    } else {
        order = o1; starts = s1; lens = l1;
        key = pred * KS + a1;    n = n1; vbase = c1;   // may be an all-invalid row
    }

    // safe = clip(key, 0, n-1)   (scalar)
    int safe = key;
    if (safe < 0)     safe = 0;
    if (safe > n - 1) safe = n - 1;

    const int left = starts[safe];           // s_load_b32
    const int cmax = (F > 0) ? (F - 1) : 0;

    // gfx1250 prefetch: each lane prefetches its pair's line, covering the
    // whole (up to 256B) segment run while the mask ALU finishes.
    {
        int pidx = left + 2 * lane;
        pidx = (pidx < 0) ? 0 : (pidx > cmax ? cmax : pidx);
        __builtin_prefetch(order + pidx, 0, 3);
    }

    int cnt = lens[safe];                    // s_load_b32
    if (cnt > MR) cnt = MR;

    const long long base = (long long)row * MR;
    int* __restrict__ out_fact  = out;
    int* __restrict__ out_valid = out + (long long)B * MR;

    if ((MR & 1) == 0) {
        // Paired path: lane owns contiguous (j, j+1); 8B NT stores, coalesced.
        for (int j = 2 * lane; j + 1 < MR; j += 64) {
            int i0 = left + j;
            int i1 = i0 + 1;
            i0 = (i0 < 0) ? 0 : (i0 > cmax ? cmax : i0);
            i1 = (i1 < 0) ? 0 : (i1 > cmax ? cmax : i1);

            v2i fv;                                    // clamped gather (L2-hit)
            fv.x = order[i0];
            fv.y = order[i1];

            v2i vv;
            vv.x = (vbase && (j     < cnt)) ? 1 : 0;
            vv.y = (vbase && (j + 1 < cnt)) ? 1 : 0;

            // Write-once output streams: NT so L2 stays dedicated to order[].
            __builtin_nontemporal_store(fv, (v2i*)(out_fact  + base + j));
            __builtin_nontemporal_store(vv, (v2i*)(out_valid + base + j));
        }
    } else {
        // General fallback for odd MR.
        for (int j = lane; j < MR; j += 32) {
            int idx = left + j;
            idx = (idx < 0) ? 0 : (idx > cmax ? cmax : idx);
            const int fv = order[idx];
            const int v  = (vbase && (j < cnt)) ? 1 : 0;
            __builtin_nontemporal_store(fv, out_fact  + base + j);
            __builtin_nontemporal_store(v,  out_valid + base + j);
        }
    }
}

extern "C" void kernel_launch(void* const* d_in, const int* in_sizes, int n_in,
                              void* d_out, int out_size, void* d_ws, size_t ws_size,
                              hipStream_t stream) {
    const int* q   = (const int*)d_in[0];
    const int* o0  = (const int*)d_in[1];
    const int* s0  = (const int*)d_in[2];
    const int* l0  = (const int*)d_in[3];
    const int* o1  = (const int*)d_in[4];
    const int* s1  = (const int*)d_in[5];
    const int* l1  = (const int*)d_in[6];
    const int* op  = (const int*)d_in[7];
    const int* sp  = (const int*)d_in[8];
    const int* lp  = (const int*)d_in[9];
    const int* mrp = (const int*)d_in[10];   // max_results, device scalar

    const int B  = in_sizes[0] / 3;
    const int F  = in_sizes[1];
    const int n0 = in_sizes[2];
    const int n1 = in_sizes[5];
    const int np = in_sizes[8];

    // 8 waves (8 queries) per 256-thread block under wave32.
    const int blocks = (B + 7) / 8;
    ArgKeyFactIndex_gather_kernel<<<blocks, 256, 0, stream>>>(
        q, o0, s0, l0, o1, s1, l1, op, sp, lp, mrp,
        (int*)d_out, B, F, n0, n1, np);
}